// MF_73856257622285
// MI455X (gfx1250) — compile-verified
//
#include <hip/hip_runtime.h>

typedef __attribute__((ext_vector_type(2))) float v2f;
typedef __attribute__((ext_vector_type(8))) float v8f;

namespace {
constexpr int kHidden        = 64;
constexpr int kTile          = 16;  // batch elements per wave (WMMA M=N=16)
constexpr int kWavesPerBlock = 8;   // 256 threads/block = 8 wave32
constexpr int kBlockThreads  = kWavesPerBlock * 32;
}

__global__ __launch_bounds__(kBlockThreads)
void mf_dot_wmma_f32(const int* __restrict__ user,
                     const int* __restrict__ item,
                     const float* __restrict__ user_weight,
                     const float* __restrict__ item_weight,
                     const float* __restrict__ user_bias,
                     const float* __restrict__ item_bias,
                     const float* __restrict__ bias,
                     float* __restrict__ out,
                     int batch)
{
  const int lane = threadIdx.x & 31;
  const int wave = blockIdx.x * kWavesPerBlock + (threadIdx.x >> 5);
  const int tile = wave * kTile;
  if (tile >= batch) return;            // wave-uniform exit: EXEC stays full below

  const int m  = lane & 15;             // tile row this lane contributes to
  const int kh = lane >> 4;             // which K-pair of the 16x16x4 fragment
  int b = tile + m;
  if (b >= batch) b = batch - 1;        // clamp keeps all 32 lanes active for WMMA

  const int   uidx  = user[b];
  const int   iidx  = item[b];
  const float ubias = user_bias[uidx];  // [U,1] bias broadcast over hidden
  const float ibias = item_bias[iidx];
  const float gbias = bias[0];

  const float* __restrict__ urow = user_weight + (size_t)uidx * kHidden;
  const float* __restrict__ irow = item_weight + (size_t)iidx * kHidden;

  // C accumulates U(16xK) x I^T(Kx16); we only need the diagonal, but the
  // chip is HBM-bound here so the extra matrix compute is free.
  v8f c = {};
#pragma unroll
  for (int k0 = 0; k0 < kHidden; k0 += 4) {
    const int kk = k0 + (kh << 1);
    v2f a  = *reinterpret_cast<const v2f*>(urow + kk);   // A: user row m, K=kk..kk+1
    v2f bt = *reinterpret_cast<const v2f*>(irow + kk);   // B: item col m, K=kk..kk+1
    a.x  += ubias;  a.y  += ubias;                       // fold per-row biases
    bt.x += ibias;  bt.y += ibias;
    // D = A x B + C   (8 args: neg_a, A, neg_b, B, c_mod, C, reuse_a, reuse_b)
    c = __builtin_amdgcn_wmma_f32_16x16x4_f32(false, a, false, bt,
                                              (short)0, c, false, false);
  }

  // Diagonal (m,m) of the 16x16 f32 C/D fragment:
  //   m in [0,8):  lane m,     VGPR m
  //   m in [8,16): lane m+16,  VGPR m-8   (i.e. lanes 24..31)
  const int j = lane & 7;
  float v = c[0];
  v = (j == 1) ? c[1] : v;
  v = (j == 2) ? c[2] : v;
  v = (j == 3) ? c[3] : v;
  v = (j == 4) ? c[4] : v;
  v = (j == 5) ? c[5] : v;
  v = (j == 6) ? c[6] : v;
  v = (j == 7) ? c[7] : v;

  const bool diagLane = (lane < 8) || (lane >= 24);
  const int  outIdx   = tile + ((lane < 8) ? lane : (lane - 16));
  if (diagLane && outIdx < batch)
    out[outIdx] = v + gbias;
}

extern "C" void kernel_launch(void* const* d_in, const int* in_sizes, int n_in,
                              void* d_out, int out_size, void* d_ws, size_t ws_size,
                              hipStream_t stream) {
  const int*   user        = (const int*)  d_in[0];
  const int*   item        = (const int*)  d_in[1];
  // d_in[2] = target (unused by forward pass)
  const float* user_weight = (const float*)d_in[3];
  const float* item_weight = (const float*)d_in[4];
  const float* user_bias   = (const float*)d_in[5];
  const float* item_bias   = (const float*)d_in[6];
  const float* bias        = (const float*)d_in[7];
  float*       out         = (float*)d_out;

  const int batch          = in_sizes[0];
  const int elemsPerBlock  = kWavesPerBlock * kTile;  // 128
  const int blocks         = (batch + elemsPerBlock - 1) / elemsPerBlock;

  mf_dot_wmma_f32<<<blocks, kBlockThreads, 0, stream>>>(
      user, item, user_weight, item_weight, user_bias, item_bias, bias,
      out, batch);
}